// PolicyNetwork_15487652069411
// MI455X (gfx1250) — compile-verified
//
#include <hip/hip_runtime.h>
#include <hip/hip_bf16.h>

typedef __attribute__((ext_vector_type(2))) float v2f;
typedef __attribute__((ext_vector_type(8))) float v8f;

static __device__ __forceinline__ v8f wmma_f32_k4(v2f a, v2f b, v8f c) {
  // D = A(16x4,f32) x B(4x16,f32) + C(16x16,f32), full fp32 precision
  return __builtin_amdgcn_wmma_f32_16x16x4_f32(
      /*neg_a=*/false, a, /*neg_b=*/false, b,
      /*c_mod=*/(short)0, c, /*reuse_a=*/false, /*reuse_b=*/false);
}

// ---------------------------------------------------------------------------
// Kernel 1: per-job bias for the node MLP:
//   pre[j][k] = dag_emb[j] @ w1n[24:40] + g @ w1n[40:56] + b1n   (J x 16)
// ---------------------------------------------------------------------------
__global__ __launch_bounds__(256) void pre_node_kernel(
    const float* __restrict__ de, const float* __restrict__ g,
    const float* __restrict__ w1n, const float* __restrict__ b1n,
    float* __restrict__ pre, int J) {
  int idx = blockIdx.x * blockDim.x + threadIdx.x;
  if (idx >= J * 16) return;
  int j = idx >> 4, k = idx & 15;
  float acc = b1n[k];
  const float* der = de + (size_t)j * 16;
#pragma unroll
  for (int e = 0; e < 16; ++e) acc = fmaf(der[e], w1n[(24 + e) * 16 + k], acc);
#pragma unroll
  for (int e = 0; e < 16; ++e) acc = fmaf(g[e], w1n[(40 + e) * 16 + k], acc);
  pre[idx] = acc;
}

// ---------------------------------------------------------------------------
// Kernel 2: main node MLP over N nodes, WMMA fp32, 16 nodes per wave32.
// ---------------------------------------------------------------------------
__global__ __launch_bounds__(256) void op_mlp_kernel(
    const float* __restrict__ nf, const float* __restrict__ ne,
    const float* __restrict__ pre, const int* __restrict__ ptr,
    const float* __restrict__ w1n, const float* __restrict__ w2n,
    const float* __restrict__ b2n, const float* __restrict__ w3n,
    const float* __restrict__ b3n, float* __restrict__ out,
    int J, long long N) {
  __shared__ float lds[8 * 272 + 8 * 16];  // 8 waves: 16x17 transpose + 16 scores

  const int lane = threadIdx.x & 31;
  const int wav  = threadIdx.x >> 5;
  // Wave-uniform tile index (readfirstlane -> SGPR) so the ptr search stays
  // scalar (SALU + s_load) and EXEC is never touched around the WMMAs.
  const int tileI = __builtin_amdgcn_readfirstlane(blockIdx.x * 8 + wav);
  const long long base = (long long)tileI * 16;
  if (base >= N) return;  // wave-uniform

  const int m  = lane & 15;   // A: row M ; B/C: column N
  const int hi = lane >> 4;   // lane-half selects K pair / M high half
  const int k2 = 2 * hi;

  // ----- B operand, layer 1: W1 = w1n rows 0..23 (nf part + ne part) -----
  v2f bw1[6];
#pragma unroll
  for (int t = 0; t < 6; ++t) {
    int k0 = 4 * t + k2;
    bw1[t].x = w1n[k0 * 16 + m];
    bw1[t].y = w1n[(k0 + 1) * 16 + m];
  }
  // ----- B operand, layer 2: w2n (16x8) zero-padded to 16x16 -----
  v2f bw2[4];
#pragma unroll
  for (int t = 0; t < 4; ++t) {
    int k0 = 4 * t + k2;
    bw2[t].x = (m < 8) ? w2n[k0 * 8 + m] : 0.0f;
    bw2[t].y = (m < 8) ? w2n[(k0 + 1) * 8 + m] : 0.0f;
  }

  // ----- A operand, layer 1: x = [nf(8) | ne(16)] for node base+m -----
  long long node = base + m;
  if (node >= N) node = N - 1;  // safe for partial last tile (stores masked)
  const float* nfp = nf + node * 8;
  const float* nep = ne + node * 16;
  v2f a[6];
  a[0] = *(const v2f*)(nfp + k2);
  a[1] = *(const v2f*)(nfp + 4 + k2);
  a[2] = *(const v2f*)(nep + k2);
  a[3] = *(const v2f*)(nep + 4 + k2);
  a[4] = *(const v2f*)(nep + 8 + k2);
  a[5] = *(const v2f*)(nep + 12 + k2);

  v8f c = {0.f, 0.f, 0.f, 0.f, 0.f, 0.f, 0.f, 0.f};
#pragma unroll
  for (int t = 0; t < 6; ++t) c = wmma_f32_k4(a[t], bw1[t], c);

  // ----- per-job bias: scalar binary searches over ptr (uniform indices) --
  const int base_i = tileI * 16;
  int lo = 0, hj = J - 1;
  while (lo < hj) {
    int mid = (lo + hj + 1) >> 1;
    if (ptr[mid] <= base_i) lo = mid; else hj = mid - 1;
  }
  const int j0 = lo;
  hj = J - 1;
  while (lo < hj) {
    int mid = (lo + hj + 1) >> 1;
    if (ptr[mid] <= base_i + 15) lo = mid; else hj = mid - 1;
  }
  const int j1 = lo;
  const float p0 = pre[(size_t)j0 * 16 + m];
  const float p1 = pre[(size_t)j1 * 16 + m];
  const int boundary = ptr[j1];

  // ----- relu(h + pre) -> LDS (row-major, stride 17 to avoid conflicts) ---
  float* L = lds + wav * 272;
#pragma unroll
  for (int r = 0; r < 8; ++r) {
    int M = r + 8 * hi;
    float p = (base_i + M >= boundary) ? p1 : p0;
    L[M * 17 + m] = fmaxf(c[r] + p, 0.0f);
  }
  asm volatile("s_wait_dscnt 0" ::: "memory");

  // ----- layer 2: (16x16) @ (16x16-padded), A re-read transposed from LDS -
  v8f c2 = {0.f, 0.f, 0.f, 0.f, 0.f, 0.f, 0.f, 0.f};
#pragma unroll
  for (int t = 0; t < 4; ++t) {
    int k0 = 4 * t + k2;
    v2f a2 = {L[m * 17 + k0], L[m * 17 + k0 + 1]};
    c2 = wmma_f32_k4(a2, bw2[t], c2);
  }

  // ----- layer 3: dot with w3n over 8 valid columns + xor-reduce ----------
  const float w3v = (m < 8) ? w3n[m] : 0.0f;
  const float b2v = (m < 8) ? b2n[m] : 0.0f;
  const float b3v = b3n[0];
  float* S = lds + 8 * 272 + wav * 16;
#pragma unroll
  for (int r = 0; r < 8; ++r) {
    float h2 = fmaxf(c2[r] + b2v, 0.0f);  // padded cols: exactly 0
    float v = h2 * w3v;
    v += __shfl_xor(v, 1, 32);
    v += __shfl_xor(v, 2, 32);
    v += __shfl_xor(v, 4, 32);           // stays inside 8-lane groups
    if (m == 0) S[r + 8 * hi] = v + b3v; // lanes 0 and 16 park scores in LDS
  }
  asm volatile("s_wait_dscnt 0" ::: "memory");
  // One coalesced 64B store per wave from lanes 0..15.
  if (hi == 0 && base + m < N) out[base + m] = S[m];
}

// ---------------------------------------------------------------------------
// Kernel 3a: per-job hidden for dag MLP:
//   hjob[j][k] = nf[ptr[j],:4] @ w1d[0:4] + de[j] @ w1d[4:20]
//              + g @ w1d[20:36] + b1d[k]
// ---------------------------------------------------------------------------
__global__ __launch_bounds__(256) void hjob_kernel(
    const float* __restrict__ nf, const float* __restrict__ de,
    const float* __restrict__ g, const int* __restrict__ ptr,
    const float* __restrict__ w1d, const float* __restrict__ b1d,
    float* __restrict__ hjob, int J) {
  int idx = blockIdx.x * blockDim.x + threadIdx.x;
  if (idx >= J * 16) return;
  int j = idx >> 4, k = idx & 15;
  float acc = b1d[k];
  const float* nfr = nf + (size_t)ptr[j] * 8;
#pragma unroll
  for (int i = 0; i < 4; ++i) acc = fmaf(nfr[i], w1d[i * 16 + k], acc);
  const float* der = de + (size_t)j * 16;
#pragma unroll
  for (int e = 0; e < 16; ++e) acc = fmaf(der[e], w1d[(4 + e) * 16 + k], acc);
#pragma unroll
  for (int e = 0; e < 16; ++e) acc = fmaf(g[e], w1d[(20 + e) * 16 + k], acc);
  hjob[idx] = acc;
}

// ---------------------------------------------------------------------------
// Kernel 3b: prlvl_scores[j][w] — 500k cheap rows, VALU.
// ---------------------------------------------------------------------------
__global__ __launch_bounds__(256) void prlvl_kernel(
    const float* __restrict__ hjob, const float* __restrict__ w1d,
    const float* __restrict__ w2d, const float* __restrict__ b2d,
    const float* __restrict__ w3d, const float* __restrict__ b3d,
    float* __restrict__ out, int J, int W) {
  int idx = blockIdx.x * blockDim.x + threadIdx.x;
  if (idx >= J * W) return;
  int j = idx / W;
  float wf = (float)(idx - j * W);
  float hd[16];
#pragma unroll
  for (int k = 0; k < 16; ++k)
    hd[k] = fmaxf(hjob[(size_t)j * 16 + k] + wf * w1d[36 * 16 + k], 0.0f);
  float o = b3d[0];
#pragma unroll
  for (int oo = 0; oo < 8; ++oo) {
    float acc = b2d[oo];
#pragma unroll
    for (int k = 0; k < 16; ++k) acc = fmaf(hd[k], w2d[k * 8 + oo], acc);
    o = fmaf(fmaxf(acc, 0.0f), w3d[oo], o);
  }
  out[idx] = o;
}

// ---------------------------------------------------------------------------
extern "C" void kernel_launch(void* const* d_in, const int* in_sizes, int n_in,
                              void* d_out, int out_size, void* d_ws, size_t ws_size,
                              hipStream_t stream) {
  const float* nf  = (const float*)d_in[0];   // (N,8)
  const float* ne  = (const float*)d_in[1];   // (N,16)
  const float* de  = (const float*)d_in[2];   // (J,16)
  const float* g   = (const float*)d_in[3];   // (16,)
  const int*   ptr = (const int*)d_in[5];     // (J+1,)
  const float* w1n = (const float*)d_in[7];
  const float* b1n = (const float*)d_in[8];
  const float* w2n = (const float*)d_in[9];
  const float* b2n = (const float*)d_in[10];
  const float* w3n = (const float*)d_in[11];
  const float* b3n = (const float*)d_in[12];
  const float* w1d = (const float*)d_in[13];
  const float* b1d = (const float*)d_in[14];
  const float* w2d = (const float*)d_in[15];
  const float* b2d = (const float*)d_in[16];
  const float* w3d = (const float*)d_in[17];
  const float* b3d = (const float*)d_in[18];

  const long long N = (long long)in_sizes[0] / 8;
  const int J = in_sizes[2] / 16;
  const int W = (int)(((long long)out_size - N) / J);

  float* pre  = (float*)d_ws;                 // J*16 floats
  float* hjob = pre + (size_t)J * 16;         // J*16 floats
  float* out  = (float*)d_out;

  const int t1 = J * 16;
  pre_node_kernel<<<(t1 + 255) / 256, 256, 0, stream>>>(de, g, w1n, b1n, pre, J);
  hjob_kernel<<<(t1 + 255) / 256, 256, 0, stream>>>(nf, de, g, ptr, w1d, b1d, hjob, J);

  const long long tiles = (N + 15) / 16;
  const int blocks = (int)((tiles + 7) / 8);  // 8 waves (tiles) per 256-thread block
  op_mlp_kernel<<<blocks, 256, 0, stream>>>(nf, ne, pre, ptr, w1n, w2n, b2n, w3n,
                                            b3n, out, J, N);

  const int t3 = J * W;
  prlvl_kernel<<<(t3 + 255) / 256, 256, 0, stream>>>(hjob, w1d, w2d, b2d, w3d,
                                                     b3d, out + N, J, W);
}